// PT_GraphVAE_23733989278150
// MI455X (gfx1250) — compile-verified
//
#include <hip/hip_runtime.h>

#define NT_   3712
#define NB_   32
#define NN_   116
#define CIN_  116
#define HID_  128
#define EMB_  64
#define DECH_ 64
#define EPOS_ 64000
#define ETOT_ 128000
#define TK_   4096

typedef __attribute__((ext_vector_type(2))) float v2f;
typedef __attribute__((ext_vector_type(8))) float v8f;

__device__ __forceinline__ v8f wmma4(v2f a, v2f b, v8f c) {
  // D = A(16x4 f32) * B(4x16 f32) + C(16x16 f32)
  return __builtin_amdgcn_wmma_f32_16x16x4_f32(false, a, false, b, (short)0, c, false, false);
}

__device__ __forceinline__ float mishf(float x) {
  float sp = (x > 20.0f) ? x : log1pf(expf(x));
  return x * tanhf(sp);
}

__device__ __forceinline__ void block_reduce_add(float v, float* target) {
  __shared__ float red[256];
  int t = threadIdx.x;
  red[t] = v;
  __syncthreads();
  for (int s = blockDim.x >> 1; s > 0; s >>= 1) {
    if (t < s) red[t] += red[t + s];
    __syncthreads();
  }
  if (t == 0) atomicAdd(target, red[0]);
}

// ---------------------------------------------------------------- utilities
__global__ void zero_kernel(float* p, long long n) {
  long long i = (long long)blockIdx.x * blockDim.x + threadIdx.x;
  if (i < n) p[i] = 0.0f;
}

__global__ void labels_kernel(float* labels) {
  int i = blockIdx.x * blockDim.x + threadIdx.x;
  if (i < ETOT_) labels[i] = (i < EPOS_) ? 1.0f : 0.0f;
}

// ------------------------------------------------ SAGE aggregation (scatter)
__global__ void enc_deg_kernel(const int* __restrict__ ps, const int* __restrict__ pd,
                               float* __restrict__ deg) {
  int e = blockIdx.x * blockDim.x + threadIdx.x;
  if (e >= ETOT_) return;  // symmetrized positive edges: 2 * EPOS_
  int d = (e < EPOS_) ? pd[e] : ps[e - EPOS_];
  atomicAdd(&deg[d], 1.0f);
}

__global__ void enc_agg_kernel(const int* __restrict__ ps, const int* __restrict__ pd,
                               const float* __restrict__ X, float* __restrict__ agg, int F) {
  long long t = (long long)blockIdx.x * blockDim.x + threadIdx.x;
  if (t >= (long long)ETOT_ * F) return;
  int e = (int)(t / F), f = (int)(t % F);
  int s, d;
  if (e < EPOS_) { s = ps[e]; d = pd[e]; }
  else           { s = pd[e - EPOS_]; d = ps[e - EPOS_]; }
  atomicAdd(&agg[(size_t)d * F + f], X[(size_t)s * F + f]);
}

__device__ __forceinline__ void dec_edge(int e, const int* ps, const int* pd,
                                         const int* ns, const int* nd2,
                                         int& s, int& d, int& base) {
  base = (e < ETOT_) ? e : e - ETOT_;
  int ss, dd;
  if (base < EPOS_) { ss = ps[base]; dd = pd[base]; }
  else              { ss = ns[base - EPOS_]; dd = nd2[base - EPOS_]; }
  if (e < ETOT_) { s = ss; d = dd; } else { s = dd; d = ss; }
}

__global__ void dec_deg_kernel(const int* __restrict__ ps, const int* __restrict__ pd,
                               const int* __restrict__ ns, const int* __restrict__ nd2,
                               const float* __restrict__ keep, float* __restrict__ deg) {
  int e = blockIdx.x * blockDim.x + threadIdx.x;
  if (e >= 2 * ETOT_) return;
  int s, d, base;
  dec_edge(e, ps, pd, ns, nd2, s, d, base);
  float w = keep[base];
  if (w != 0.0f) atomicAdd(&deg[d], w);
}

__global__ void dec_agg_kernel(const int* __restrict__ ps, const int* __restrict__ pd,
                               const int* __restrict__ ns, const int* __restrict__ nd2,
                               const float* __restrict__ keep,
                               const float* __restrict__ Z, float* __restrict__ agg) {
  long long t = (long long)blockIdx.x * blockDim.x + threadIdx.x;
  if (t >= (long long)2 * ETOT_ * EMB_) return;
  int e = (int)(t / EMB_), f = (int)(t % EMB_);
  int s, d, base;
  dec_edge(e, ps, pd, ns, nd2, s, d, base);
  float w = keep[base];
  if (w != 0.0f) atomicAdd(&agg[(size_t)d * EMB_ + f], Z[(size_t)s * EMB_ + f] * w);
}

__global__ void div_deg_kernel(const float* __restrict__ deg, float* __restrict__ agg,
                               long long total, int F) {
  long long t = (long long)blockIdx.x * blockDim.x + threadIdx.x;
  if (t >= total) return;
  agg[t] /= fmaxf(deg[t / F], 1.0f);
}

// ------------------------------------------- WMMA fp32 GEMM: Y = act(A1*W1^T + b [+ A2*W2^T])
// A [M,K] row-major, W [N,K] row-major, Y [M,N]. M%16==0, N%16==0, K%4==0, K even.
// DUAL / ACT are compile-time so the inner loop is branch-free; fragments are
// register double-buffered so next iteration's loads overlap the WMMA.
template <bool DUAL, int ACT>
__global__ void wmma_lin_kernel(const float* __restrict__ A1, const float* __restrict__ W1,
                                const float* __restrict__ b1,
                                const float* __restrict__ A2, const float* __restrict__ W2,
                                float* __restrict__ Y, int M, int K, int N) {
  int wave = (int)(((long long)blockIdx.x * blockDim.x + threadIdx.x) >> 5);
  int ntile = N >> 4;
  int total = (M >> 4) * ntile;
  if (wave >= total) return;
  int lane = threadIdx.x & 31;
  int half = lane >> 4, lrow = lane & 15;
  int mt = wave / ntile, nt = wave - mt * ntile;
  const float* a1 = A1 + (size_t)(mt * 16 + lrow) * K;
  const float* w1 = W1 + (size_t)(nt * 16 + lrow) * K;
  const float* a2 = DUAL ? A2 + (size_t)(mt * 16 + lrow) * K : a1;
  const float* w2 = DUAL ? W2 + (size_t)(nt * 16 + lrow) * K : w1;
  v8f c = {0.f, 0.f, 0.f, 0.f, 0.f, 0.f, 0.f, 0.f};

  int j = 2 * half;
  v2f a1c, b1c, a2c, b2c;
  a1c = *(const v2f*)(a1 + j);
  b1c = *(const v2f*)(w1 + j);
  if (DUAL) {
    a2c = *(const v2f*)(a2 + j);
    b2c = *(const v2f*)(w2 + j);
  }
  for (int k = 4; k < K; k += 4) {
    int jn = k + 2 * half;
    v2f a1n, b1n, a2n, b2n;
    a1n = *(const v2f*)(a1 + jn);
    b1n = *(const v2f*)(w1 + jn);
    if (DUAL) {
      a2n = *(const v2f*)(a2 + jn);
      b2n = *(const v2f*)(w2 + jn);
    }
    c = wmma4(a1c, b1c, c);
    if (DUAL) c = wmma4(a2c, b2c, c);
    a1c = a1n; b1c = b1n;
    if (DUAL) { a2c = a2n; b2c = b2n; }
  }
  c = wmma4(a1c, b1c, c);
  if (DUAL) c = wmma4(a2c, b2c, c);

  int col = nt * 16 + lrow;
  float bias = b1 ? b1[col] : 0.0f;
#pragma unroll
  for (int r = 0; r < 8; ++r) {
    int row = mt * 16 + r + 8 * half;
    float v = c[r] + bias;
    if (ACT) v = mishf(v);
    Y[(size_t)row * N + col] = v;
  }
}

// -------------------------------------------------- reparameterization, KL
__global__ void reparam_kernel(const float* __restrict__ mu, const float* __restrict__ lv,
                               const float* __restrict__ eps, float* __restrict__ z, int n) {
  int i = blockIdx.x * blockDim.x + threadIdx.x;
  if (i < n) z[i] = mu[i] + eps[i] * expf(0.5f * lv[i]);
}

__global__ void kl_kernel(const float* __restrict__ mu, const float* __restrict__ lv,
                          float* acc, int n) {
  int i = blockIdx.x * blockDim.x + threadIdx.x;
  float v = 0.0f;
  if (i < n) {
    float l = lv[i], m = mu[i];
    v = 1.0f + l - m * m - expf(l);
  }
  block_reduce_add(v, acc);
}

// -------------------------------------------------------- fused edge decoder
// feat = [s+d (K 0..63) | |s-d| (64..127) | s*d (128..191)] -> three branch-free
// 64-K segments; each segment step: 2x global_load_b64 (A pair) + 4x b64 (W) + 4 WMMA.
template <int OP>
__device__ __forceinline__ void edge_seg(const float* __restrict__ zs, const float* __restrict__ zd,
                                         const float* __restrict__ w0, const float* __restrict__ w1,
                                         const float* __restrict__ w2, const float* __restrict__ w3,
                                         int half, int kbase, v8f acc[4]) {
  for (int k = 0; k < 64; k += 4) {
    int j = k + 2 * half;
    v2f sv = *(const v2f*)(zs + j);
    v2f dv = *(const v2f*)(zd + j);
    v2f av;
    if (OP == 0)      { av.x = sv.x + dv.x;          av.y = sv.y + dv.y; }
    else if (OP == 1) { av.x = fabsf(sv.x - dv.x);   av.y = fabsf(sv.y - dv.y); }
    else              { av.x = sv.x * dv.x;          av.y = sv.y * dv.y; }
    int jj = kbase + j;
    acc[0] = wmma4(av, *(const v2f*)(w0 + jj), acc[0]);
    acc[1] = wmma4(av, *(const v2f*)(w1 + jj), acc[1]);
    acc[2] = wmma4(av, *(const v2f*)(w2 + jj), acc[2]);
    acc[3] = wmma4(av, *(const v2f*)(w3 + jj), acc[3]);
  }
}

__global__ void edge_decode_kernel(const float* __restrict__ z,
                                   const int* __restrict__ ps, const int* __restrict__ pd,
                                   const int* __restrict__ ns, const int* __restrict__ nd2,
                                   const float* __restrict__ fc1w, const float* __restrict__ fc1b,
                                   const float* __restrict__ lng, const float* __restrict__ lnb,
                                   const float* __restrict__ fc2w, const float* __restrict__ fc2b,
                                   float* __restrict__ logits) {
  __shared__ float tile[4][16][68];
  int w = threadIdx.x >> 5;
  int lane = threadIdx.x & 31;
  int half = lane >> 4, lrow = lane & 15;
  int wave = blockIdx.x * 4 + w;
  int e = wave * 16 + lrow;
  int s, d;
  if (e < EPOS_) { s = ps[e]; d = pd[e]; }
  else           { s = ns[e - EPOS_]; d = nd2[e - EPOS_]; }
  const float* zs = z + (size_t)s * EMB_;
  const float* zd = z + (size_t)d * EMB_;
  const float* w0 = fc1w + (size_t)(0 * 16 + lrow) * 192;
  const float* w1 = fc1w + (size_t)(1 * 16 + lrow) * 192;
  const float* w2 = fc1w + (size_t)(2 * 16 + lrow) * 192;
  const float* w3 = fc1w + (size_t)(3 * 16 + lrow) * 192;

  v8f vzero = {0.f, 0.f, 0.f, 0.f, 0.f, 0.f, 0.f, 0.f};
  v8f acc[4] = {vzero, vzero, vzero, vzero};
  edge_seg<0>(zs, zd, w0, w1, w2, w3, half, 0, acc);
  edge_seg<1>(zs, zd, w0, w1, w2, w3, half, 64, acc);
  edge_seg<2>(zs, zd, w0, w1, w2, w3, half, 128, acc);

#pragma unroll
  for (int t = 0; t < 4; ++t) {
    float bias = fc1b[t * 16 + lrow];
#pragma unroll
    for (int r = 0; r < 8; ++r)
      tile[w][r + 8 * half][t * 16 + lrow] = acc[t][r] + bias;
  }
  __syncthreads();
  if (lane < 16) {
    float* rowp = tile[w][lane];
    float m = 0.0f;
    for (int ci = 0; ci < 64; ++ci) m += rowp[ci];
    m *= (1.0f / 64.0f);
    float var = 0.0f;
    for (int ci = 0; ci < 64; ++ci) { float dd = rowp[ci] - m; var += dd * dd; }
    var *= (1.0f / 64.0f);
    float inv = rsqrtf(var + 1e-5f);
    float lsum = 0.0f;
    for (int ci = 0; ci < 64; ++ci) {
      float hv = (rowp[ci] - m) * inv * lng[ci] + lnb[ci];
      lsum += mishf(hv) * fc2w[ci];
    }
    logits[wave * 16 + lane] = lsum + fc2b[0];
  }
}

// ------------------------------------------------------------ BCE reduction
__global__ void bce_kernel(const float* __restrict__ logits, float* acc) {
  int i = blockIdx.x * blockDim.x + threadIdx.x;
  float v = 0.0f;
  if (i < ETOT_) {
    float l = logits[i];
    float y = (i < EPOS_) ? 1.0f : 0.0f;
    v = fmaxf(l, 0.0f) - l * y + log1pf(expf(-fabsf(l)));
  }
  block_reduce_add(v, acc);
}

// ----------------------------------------- per-graph top-k via bitonic sort
__global__ void topk_kernel(const float* __restrict__ logits, float* __restrict__ keep) {
  __shared__ float key[TK_];
  __shared__ int idx[TK_];
  int g = blockIdx.x, tid = threadIdx.x;
  const int T = 256;
  for (int i = tid; i < TK_; i += T) {
    if (i < 4000) {
      int e = (i < 2000) ? (g * 2000 + i) : (EPOS_ + g * 2000 + (i - 2000));
      float l = logits[e];
      key[i] = 1.0f / (1.0f + expf(-l));
      idx[i] = e;
    } else {
      key[i] = -1.0f;
      idx[i] = -1;
    }
  }
  __syncthreads();
  for (int ksz = 2; ksz <= TK_; ksz <<= 1) {
    for (int j = ksz >> 1; j > 0; j >>= 1) {
      for (int i = tid; i < TK_; i += T) {
        int ixj = i ^ j;
        if (ixj > i) {
          bool descend = ((i & ksz) == 0);
          float a = key[i], b = key[ixj];
          bool sw = descend ? (a < b) : (a > b);
          if (sw) {
            key[i] = b; key[ixj] = a;
            int t2 = idx[i]; idx[i] = idx[ixj]; idx[ixj] = t2;
          }
        }
      }
      __syncthreads();
    }
  }
  for (int i = tid; i < 4000; i += T)
    keep[idx[i]] = (i < 2400) ? 1.0f : 0.0f;  // k = floor(0.6f * 4000) = 2400
}

// -------------------------------------- symmetrize + diag + MSE, final loss
__global__ void sym_mse_kernel(const float* __restrict__ raw, const float* __restrict__ orig,
                               float* __restrict__ xr, float* acc) {
  int t = blockIdx.x * blockDim.x + threadIdx.x;
  float v2 = 0.0f;
  if (t < NB_ * NN_ * NN_) {
    int b = t / (NN_ * NN_);
    int r = (t / NN_) % NN_;
    int c = t % NN_;
    float v;
    if (r == c) v = 1.0f;
    else v = 0.5f * (raw[b * NN_ * NN_ + r * NN_ + c] + raw[b * NN_ * NN_ + c * NN_ + r]);
    xr[t] = v;
    float dd = v - orig[t];
    v2 = dd * dd;
  }
  block_reduce_add(v2, acc);
}

__global__ void combine_kernel(const float* sc, float* out) {
  if (threadIdx.x == 0 && blockIdx.x == 0) {
    float bce = sc[0] / (float)ETOT_;
    float mse = sc[1] / (float)(NT_ * NN_);
    float kln = sc[2] / (float)(NT_ * EMB_);
    float kle = sc[3] / (float)(NT_ * EMB_);
    float kl = (-0.5f * kle - 0.5f * kln) * 0.5f;
    out[0] = bce + mse + 0.1f * kl;
  }
}

// ---------------------------------------------------------------- launcher
static inline int cdiv(long long a, long long b) { return (int)((a + b - 1) / b); }

static void launch_lin(const float* A1, const float* W1, const float* b1,
                       const float* A2, const float* W2, float* Y,
                       int M, int K, int N, int act, hipStream_t s) {
  long long waves = (long long)(M / 16) * (N / 16);
  int blocks = cdiv(waves * 32, 128);
  if (A2) {
    if (act) wmma_lin_kernel<true, 1><<<blocks, 128, 0, s>>>(A1, W1, b1, A2, W2, Y, M, K, N);
    else     wmma_lin_kernel<true, 0><<<blocks, 128, 0, s>>>(A1, W1, b1, A2, W2, Y, M, K, N);
  } else {
    if (act) wmma_lin_kernel<false, 1><<<blocks, 128, 0, s>>>(A1, W1, b1, A1, W1, Y, M, K, N);
    else     wmma_lin_kernel<false, 0><<<blocks, 128, 0, s>>>(A1, W1, b1, A1, W1, Y, M, K, N);
  }
}

extern "C" void kernel_launch(void* const* d_in, const int* in_sizes, int n_in,
                              void* d_out, int out_size, void* d_ws, size_t ws_size,
                              hipStream_t stream) {
  (void)in_sizes; (void)out_size; (void)ws_size;
  if (n_in < 39) return;

  const float* x        = (const float*)d_in[0];
  const int*   pos      = (const int*)d_in[1];
  const int*   neg      = (const int*)d_in[2];
  const float* orig     = (const float*)d_in[3];
  // d_in[4] = batch (implicit in edge layout)
  const float* eps_node = (const float*)d_in[5];
  const float* eps_edge = (const float*)d_in[6];
  const int* ps = pos;            const int* pd  = pos + EPOS_;
  const int* ns = neg;            const int* nd2 = neg + EPOS_;
  const float* e1_wl = (const float*)d_in[7];
  const float* e1_bl = (const float*)d_in[8];
  const float* e1_wr = (const float*)d_in[9];
  const float* e2_wl = (const float*)d_in[10];
  const float* e2_bl = (const float*)d_in[11];
  const float* e2_wr = (const float*)d_in[12];
  const float* nmu_w = (const float*)d_in[13];
  const float* nmu_b = (const float*)d_in[14];
  const float* nlv_w = (const float*)d_in[15];
  const float* nlv_b = (const float*)d_in[16];
  const float* ep_w  = (const float*)d_in[17];
  const float* ep_b  = (const float*)d_in[18];
  const float* emu_w = (const float*)d_in[19];
  const float* emu_b = (const float*)d_in[20];
  const float* elv_w = (const float*)d_in[21];
  const float* elv_b = (const float*)d_in[22];
  const float* dfc1w = (const float*)d_in[23];
  const float* dfc1b = (const float*)d_in[24];
  const float* dlng  = (const float*)d_in[25];
  const float* dlnb  = (const float*)d_in[26];
  const float* dfc2w = (const float*)d_in[27];
  const float* dfc2b = (const float*)d_in[28];
  const float* c1wl  = (const float*)d_in[29];
  const float* c1bl  = (const float*)d_in[30];
  const float* c1wr  = (const float*)d_in[31];
  const float* c2wl  = (const float*)d_in[32];
  const float* c2bl  = (const float*)d_in[33];
  const float* c2wr  = (const float*)d_in[34];
  const float* f1w   = (const float*)d_in[35];
  const float* f1b   = (const float*)d_in[36];
  const float* f2w   = (const float*)d_in[37];
  const float* f2b   = (const float*)d_in[38];

  float* out    = (float*)d_out;
  float* logits = out + 1;
  float* labels = out + 1 + ETOT_;
  float* xr     = out + 1 + 2 * ETOT_;

  float* W = (float*)d_ws;
  size_t o = 0;
  float* deg    = W + o; o += NT_;
  float* agg    = W + o; o += (size_t)NT_ * HID_;
  float* h1     = W + o; o += (size_t)NT_ * HID_;
  float* h2     = W + o; o += (size_t)NT_ * HID_;
  float* nmu    = W + o; o += (size_t)NT_ * EMB_;
  float* nlv    = W + o; o += (size_t)NT_ * EMB_;
  float* eh     = W + o; o += (size_t)NT_ * EMB_;
  float* emu    = W + o; o += (size_t)NT_ * EMB_;
  float* elv    = W + o; o += (size_t)NT_ * EMB_;
  float* znode  = W + o; o += (size_t)NT_ * EMB_;
  float* zedge  = W + o; o += (size_t)NT_ * EMB_;
  float* keep   = W + o; o += ETOT_;
  float* z1     = W + o; o += (size_t)NT_ * EMB_;
  float* z2     = W + o; o += (size_t)NT_ * EMB_;
  float* hidden = W + o; o += (size_t)NB_ * NN_ * EMB_;
  float* raw    = W + o; o += (size_t)NB_ * NN_ * NN_;
  float* sc     = W + o; o += 4;

  zero_kernel<<<1, 256, 0, stream>>>(sc, 4);

  // ---- encoder: degree (same for both layers)
  zero_kernel<<<cdiv(NT_, 256), 256, 0, stream>>>(deg, NT_);
  enc_deg_kernel<<<cdiv(ETOT_, 256), 256, 0, stream>>>(ps, pd, deg);

  // ---- SAGE layer 1 (116 -> 128) + mish
  zero_kernel<<<cdiv((long long)NT_ * CIN_, 256), 256, 0, stream>>>(agg, (long long)NT_ * CIN_);
  enc_agg_kernel<<<cdiv((long long)ETOT_ * CIN_, 256), 256, 0, stream>>>(ps, pd, x, agg, CIN_);
  div_deg_kernel<<<cdiv((long long)NT_ * CIN_, 256), 256, 0, stream>>>(deg, agg, (long long)NT_ * CIN_, CIN_);
  launch_lin(agg, e1_wl, e1_bl, x, e1_wr, h1, NT_, CIN_, HID_, 1, stream);

  // ---- SAGE layer 2 (128 -> 128) + mish
  zero_kernel<<<cdiv((long long)NT_ * HID_, 256), 256, 0, stream>>>(agg, (long long)NT_ * HID_);
  enc_agg_kernel<<<cdiv((long long)ETOT_ * HID_, 256), 256, 0, stream>>>(ps, pd, h1, agg, HID_);
  div_deg_kernel<<<cdiv((long long)NT_ * HID_, 256), 256, 0, stream>>>(deg, agg, (long long)NT_ * HID_, HID_);
  launch_lin(agg, e2_wl, e2_bl, h1, e2_wr, h2, NT_, HID_, HID_, 1, stream);

  // ---- VAE heads
  launch_lin(h2, nmu_w, nmu_b, 0, 0, nmu, NT_, HID_, EMB_, 0, stream);
  launch_lin(h2, nlv_w, nlv_b, 0, 0, nlv, NT_, HID_, EMB_, 0, stream);
  launch_lin(h2, ep_w, ep_b, 0, 0, eh, NT_, HID_, EMB_, 1, stream);
  launch_lin(eh, emu_w, emu_b, 0, 0, emu, NT_, EMB_, EMB_, 0, stream);
  launch_lin(eh, elv_w, elv_b, 0, 0, elv, NT_, EMB_, EMB_, 0, stream);
  reparam_kernel<<<cdiv(NT_ * EMB_, 256), 256, 0, stream>>>(nmu, nlv, eps_node, znode, NT_ * EMB_);
  reparam_kernel<<<cdiv(NT_ * EMB_, 256), 256, 0, stream>>>(emu, elv, eps_edge, zedge, NT_ * EMB_);

  // ---- edge decoder (fused WMMA + LN + mish + fc2): 128000 logits
  edge_decode_kernel<<<ETOT_ / 64, 128, 0, stream>>>(zedge, ps, pd, ns, nd2,
                                                     dfc1w, dfc1b, dlng, dlnb, dfc2w, dfc2b, logits);
  labels_kernel<<<cdiv(ETOT_, 256), 256, 0, stream>>>(labels);
  bce_kernel<<<cdiv(ETOT_, 256), 256, 0, stream>>>(logits, sc + 0);
  kl_kernel<<<cdiv(NT_ * EMB_, 256), 256, 0, stream>>>(nmu, nlv, sc + 2, NT_ * EMB_);
  kl_kernel<<<cdiv(NT_ * EMB_, 256), 256, 0, stream>>>(emu, elv, sc + 3, NT_ * EMB_);

  // ---- per-graph top-k keep mask
  topk_kernel<<<NB_, 256, 0, stream>>>(logits, keep);

  // ---- decoder: masked degree (same for both convs)
  zero_kernel<<<cdiv(NT_, 256), 256, 0, stream>>>(deg, NT_);
  dec_deg_kernel<<<cdiv(2 * ETOT_, 256), 256, 0, stream>>>(ps, pd, ns, nd2, keep, deg);

  // ---- decoder conv1 (64 -> 64) + mish
  zero_kernel<<<cdiv((long long)NT_ * EMB_, 256), 256, 0, stream>>>(agg, (long long)NT_ * EMB_);
  dec_agg_kernel<<<cdiv((long long)2 * ETOT_ * EMB_, 256), 256, 0, stream>>>(ps, pd, ns, nd2, keep, znode, agg);
  div_deg_kernel<<<cdiv((long long)NT_ * EMB_, 256), 256, 0, stream>>>(deg, agg, (long long)NT_ * EMB_, EMB_);
  launch_lin(agg, c1wl, c1bl, znode, c1wr, z1, NT_, EMB_, EMB_, 1, stream);

  // ---- decoder conv2 (64 -> 64) + mish
  zero_kernel<<<cdiv((long long)NT_ * EMB_, 256), 256, 0, stream>>>(agg, (long long)NT_ * EMB_);
  dec_agg_kernel<<<cdiv((long long)2 * ETOT_ * EMB_, 256), 256, 0, stream>>>(ps, pd, ns, nd2, keep, z1, agg);
  div_deg_kernel<<<cdiv((long long)NT_ * EMB_, 256), 256, 0, stream>>>(deg, agg, (long long)NT_ * EMB_, EMB_);
  launch_lin(agg, c2wl, c2bl, z1, c2wr, z2, NT_, EMB_, EMB_, 1, stream);

  // ---- big dense decoders (weight-bandwidth bound; fp32 WMMA)
  launch_lin(z2, f1w, f1b, 0, 0, hidden, NB_, NN_ * EMB_, NN_ * DECH_, 1, stream);
  launch_lin(hidden, f2w, f2b, 0, 0, raw, NB_, NN_ * DECH_, NN_ * NN_, 0, stream);

  // ---- symmetrize + unit diag + MSE; combine losses
  sym_mse_kernel<<<cdiv(NB_ * NN_ * NN_, 256), 256, 0, stream>>>(raw, orig, xr, sc + 1);
  combine_kernel<<<1, 1, 0, stream>>>(sc, out);
}